// GAT_26723286515822
// MI455X (gfx1250) — compile-verified
//
#include <hip/hip_runtime.h>
#include <hip/hip_bf16.h>
#include <math.h>

typedef float v2f __attribute__((ext_vector_type(2)));
typedef float v8f __attribute__((ext_vector_type(8)));

#define NEG_SLOPE 0.2f

// ---------------------------------------------------------------------------
// order-preserving float <-> uint mapping for atomicMax-based segment max
// ---------------------------------------------------------------------------
__device__ __forceinline__ unsigned f2key(float x) {
  unsigned b = __float_as_uint(x);
  return (b & 0x80000000u) ? ~b : (b | 0x80000000u);
}
__device__ __forceinline__ float key2f(unsigned k) {
  unsigned b = (k & 0x80000000u) ? (k & 0x7fffffffu) : ~k;
  return __uint_as_float(b);
}
__device__ __forceinline__ float lrelu(float v) {
  return v > 0.f ? v : NEG_SLOPE * v;
}

// ---------------------------------------------------------------------------
// fp32 WMMA GEMM: C[M x NC] = A[M x K] * B[K x NC]
// One wave per 16-row block, NT = NC/16 column tiles held in registers.
// A-frag (16x4 f32): lane L -> m = L%16, k = 4*step + 2*(L/16) + vgpr
// B-frag (4x16 f32): lane L -> n = L%16, k mirrored
// C/D (16x16 f32):   vgpr v, lanes 0-15 -> M=v, lanes 16-31 -> M=v+8
// ---------------------------------------------------------------------------
template <int K, int NC, int NT>
__global__ void gemm_wmma_f32(const float* __restrict__ A,
                              const float* __restrict__ B,
                              float* __restrict__ C, int M) {
  const int wave = (int)((blockIdx.x * blockDim.x + threadIdx.x) >> 5);
  const int lane = (int)(threadIdx.x & 31u);
  const int nRowBlk = M >> 4;
  if (wave >= nRowBlk) return;  // uniform per wave: EXEC stays all-ones
  const int row0 = wave << 4;
  const int m = lane & 15;
  const int kq = lane >> 4;

  v8f acc[NT] = {};
  const float* __restrict__ arow = A + (size_t)(row0 + m) * K;

  for (int kb = 0; kb < K; kb += 4) {
    const int k0 = kb + 2 * kq;
    v2f a;
    a.x = arow[k0];
    a.y = arow[k0 + 1];
    const float* __restrict__ bp = B + (size_t)k0 * NC + m;
#pragma unroll
    for (int t = 0; t < NT; ++t) {
      v2f b;
      b.x = bp[t * 16];
      b.y = bp[t * 16 + NC];
      acc[t] = __builtin_amdgcn_wmma_f32_16x16x4_f32(
          false, a, false, b, (short)0, acc[t], false, false);
    }
  }

  const int rbase = row0 + (kq << 3);
#pragma unroll
  for (int t = 0; t < NT; ++t) {
    const int col = t * 16 + m;
#pragma unroll
    for (int v = 0; v < 8; ++v)
      C[(size_t)(rbase + v) * NC + col] = acc[t][v];
  }
}

// scalar tail for rows not covered by 16-row WMMA blocks (launched only if any)
__global__ void gemm_tail(const float* __restrict__ A, const float* __restrict__ B,
                          float* __restrict__ C, int M0, int M, int K, int NC) {
  int idx = blockIdx.x * blockDim.x + threadIdx.x;
  int total = (M - M0) * NC;
  if (idx >= total) return;
  int r = M0 + idx / NC, c = idx % NC;
  float s = 0.f;
  for (int k = 0; k < K; ++k) s += A[(size_t)r * K + k] * B[(size_t)k * NC + c];
  C[(size_t)r * NC + c] = s;
}

// ---------------------------------------------------------------------------
// attention dot products: a_s[n,h] = sum_c h[n,h,c]*att_s[h,c], same for a_d
// ---------------------------------------------------------------------------
__global__ void attn_dots(const float* __restrict__ H, const float* __restrict__ att_s,
                          const float* __restrict__ att_d, float* __restrict__ as,
                          float* __restrict__ ad, int N, int heads, int C, int ldh) {
  int n = blockIdx.x * blockDim.x + threadIdx.x;
  if (n >= N) return;
  const float* row = H + (size_t)n * ldh;
  for (int h = 0; h < heads; ++h) {
    float s = 0.f, d = 0.f;
    for (int c = 0; c < C; ++c) {
      float v = row[h * C + c];
      s += v * att_s[h * C + c];
      d += v * att_d[h * C + c];
    }
    as[n * heads + h] = s;
    ad[n * heads + h] = d;
  }
}

// ---------------------------------------------------------------------------
// edge passes for segment softmax + aggregation
// ---------------------------------------------------------------------------
__global__ void edge_max(const int* __restrict__ src, const int* __restrict__ dst,
                         const float* __restrict__ as, const float* __restrict__ ad,
                         unsigned* __restrict__ mkey, int E, int heads) {
  int e = blockIdx.x * blockDim.x + threadIdx.x;
  if (e >= E) return;
  int s = src[e], d = dst[e];
  for (int h = 0; h < heads; ++h) {
    float v = lrelu(as[s * heads + h] + ad[d * heads + h]);
    atomicMax(&mkey[d * heads + h], f2key(v));
  }
}

__global__ void edge_expsum(const int* __restrict__ src, const int* __restrict__ dst,
                            const float* __restrict__ as, const float* __restrict__ ad,
                            const unsigned* __restrict__ mkey, float* __restrict__ den,
                            int E, int heads) {
  int e = blockIdx.x * blockDim.x + threadIdx.x;
  if (e >= E) return;
  int s = src[e], d = dst[e];
  for (int h = 0; h < heads; ++h) {
    float v = lrelu(as[s * heads + h] + ad[d * heads + h]);
    float m = key2f(mkey[d * heads + h]);
    if (!isfinite(m)) m = 0.f;
    atomicAdd(&den[d * heads + h], __expf(v - m));
  }
}

__global__ void edge_agg(const int* __restrict__ src, const int* __restrict__ dst,
                         const float* __restrict__ as, const float* __restrict__ ad,
                         const unsigned* __restrict__ mkey, const float* __restrict__ den,
                         const float* __restrict__ H, float* __restrict__ out,
                         int E, int heads, int C, int ldh, int ldo) {
  int e = blockIdx.x * blockDim.x + threadIdx.x;
  if (e >= E) return;
  int s = src[e], d = dst[e];
  if (e + 1 < E) __builtin_prefetch(H + (size_t)src[e + 1] * ldh, 0, 0);
  const float* hrow = H + (size_t)s * ldh;
  float* orow = out + (size_t)d * ldo;
  for (int h = 0; h < heads; ++h) {
    float v = lrelu(as[s * heads + h] + ad[d * heads + h]);
    float m = key2f(mkey[d * heads + h]);
    if (!isfinite(m)) m = 0.f;
    float alpha = __expf(v - m) / (den[d * heads + h] + 1e-16f);
    for (int c = 0; c < C; ++c)
      atomicAdd(&orow[h * C + c], hrow[h * C + c] * alpha);
  }
}

// ---------------------------------------------------------------------------
// pointwise helpers
// ---------------------------------------------------------------------------
__global__ void fill0(float* __restrict__ p, size_t n) {
  size_t i = (size_t)blockIdx.x * blockDim.x + threadIdx.x;
  size_t stride = (size_t)gridDim.x * blockDim.x;
  for (; i < n; i += stride) p[i] = 0.f;
}

__global__ void elu_bias(float* __restrict__ x, const float* __restrict__ b,
                         int total, int NC) {
  int i = blockIdx.x * blockDim.x + threadIdx.x;
  if (i >= total) return;
  float v = x[i] + b[i % NC];
  x[i] = v > 0.f ? v : (__expf(v) - 1.f);
}

__global__ void add_bias(float* __restrict__ x, const float* __restrict__ b,
                         int total, int NC) {
  int i = blockIdx.x * blockDim.x + threadIdx.x;
  if (i >= total) return;
  x[i] += b[i % NC];
}

__global__ void pad_w2(const float* __restrict__ W, float* __restrict__ Wp,
                       int K, int NCin, int NCout) {
  int i = blockIdx.x * blockDim.x + threadIdx.x;
  int total = K * NCout;
  if (i >= total) return;
  int r = i / NCout, c = i % NCout;
  Wp[i] = (c < NCin) ? W[r * NCin + c] : 0.f;
}

// ---------------------------------------------------------------------------
// launch
// ---------------------------------------------------------------------------
static inline int cdiv(long long a, long long b) { return (int)((a + b - 1) / b); }

extern "C" void kernel_launch(void* const* d_in, const int* in_sizes, int n_in,
                              void* d_out, int out_size, void* d_ws, size_t ws_size,
                              hipStream_t stream) {
  const float* x    = (const float*)d_in[0];
  const int*   ei   = (const int*)d_in[1];
  const float* W1   = (const float*)d_in[2];
  const float* aS1  = (const float*)d_in[3];
  const float* aD1  = (const float*)d_in[4];
  const float* b1   = (const float*)d_in[5];
  const float* W2   = (const float*)d_in[6];
  const float* aS2  = (const float*)d_in[7];
  const float* aD2  = (const float*)d_in[8];
  const float* b2   = (const float*)d_in[9];
  float* out = (float*)d_out;

  const int IN_C = 128, H1C = 64, HID = 32, HEADS = 2, OUT_C = 40, H2P = 48;
  const int N = in_sizes[0] / IN_C;
  const int E = in_sizes[1] / 2;
  const int* src = ei;
  const int* dst = ei + E;

  // workspace layout (floats)
  float* ws = (float*)d_ws;
  size_t off = 0;
  float*    h1   = ws + off; off += (size_t)N * H1C;
  float*    x2   = ws + off; off += (size_t)N * H1C;   // layer-1 agg, then ELU'd input
  float*    h2p  = ws + off; off += (size_t)N * H2P;   // layer-2 features, padded to 48
  float*    as1  = ws + off; off += (size_t)N * HEADS;
  float*    ad1  = ws + off; off += (size_t)N * HEADS;
  unsigned* m1   = (unsigned*)(ws + off); off += (size_t)N * HEADS;
  float*    den1 = ws + off; off += (size_t)N * HEADS;
  float*    as2  = ws + off; off += (size_t)N;
  float*    ad2  = ws + off; off += (size_t)N;
  unsigned* m2   = (unsigned*)(ws + off); off += (size_t)N;
  float*    den2 = ws + off; off += (size_t)N;
  float*    W2p  = ws + off; off += (size_t)H1C * H2P;
  (void)ws_size; (void)n_in;

  const int TB = 256;
  const int egrid = cdiv(E, TB);
  const int ngrid = cdiv(N, TB);
  auto zgrid = [&](size_t n) { int g = cdiv((long long)n, TB); return g > 4096 ? 4096 : g; };

  // --- zero accumulators / softmax state / output (deterministic per call) ---
  fill0<<<zgrid((size_t)N * H1C), TB, 0, stream>>>(x2, (size_t)N * H1C);
  fill0<<<zgrid((size_t)N * HEADS), TB, 0, stream>>>((float*)m1, (size_t)N * HEADS);
  fill0<<<zgrid((size_t)N * HEADS), TB, 0, stream>>>(den1, (size_t)N * HEADS);
  fill0<<<zgrid((size_t)N), TB, 0, stream>>>((float*)m2, (size_t)N);
  fill0<<<zgrid((size_t)N), TB, 0, stream>>>(den2, (size_t)N);
  fill0<<<zgrid((size_t)N * OUT_C), TB, 0, stream>>>(out, (size_t)N * OUT_C);

  // --- layer 1: h1 = x @ W1 (fp32 WMMA) ---
  {
    const int M16 = (N / 16) * 16;
    const int waves = N / 16;
    if (waves > 0)
      gemm_wmma_f32<128, 64, 4><<<cdiv((long long)waves * 32, TB), TB, 0, stream>>>(x, W1, h1, N);
    const int rem = N - M16;
    if (rem > 0)
      gemm_tail<<<cdiv((long long)rem * H1C, TB), TB, 0, stream>>>(x, W1, h1, M16, N, IN_C, H1C);
  }
  attn_dots<<<ngrid, TB, 0, stream>>>(h1, aS1, aD1, as1, ad1, N, HEADS, HID, H1C);
  edge_max<<<egrid, TB, 0, stream>>>(src, dst, as1, ad1, m1, E, HEADS);
  edge_expsum<<<egrid, TB, 0, stream>>>(src, dst, as1, ad1, m1, den1, E, HEADS);
  edge_agg<<<egrid, TB, 0, stream>>>(src, dst, as1, ad1, m1, den1, h1, x2,
                                     E, HEADS, HID, H1C, H1C);
  elu_bias<<<cdiv((long long)N * H1C, TB), TB, 0, stream>>>(x2, b1, N * H1C, H1C);

  // --- layer 2: h2 = elu_out @ W2 (W2 zero-padded 40 -> 48 cols) ---
  pad_w2<<<cdiv(H1C * H2P, TB), TB, 0, stream>>>(W2, W2p, H1C, OUT_C, H2P);
  {
    const int M16 = (N / 16) * 16;
    const int waves = N / 16;
    if (waves > 0)
      gemm_wmma_f32<64, 48, 3><<<cdiv((long long)waves * 32, TB), TB, 0, stream>>>(x2, W2p, h2p, N);
    const int rem = N - M16;
    if (rem > 0)
      gemm_tail<<<cdiv((long long)rem * H2P, TB), TB, 0, stream>>>(x2, W2p, h2p, M16, N, H1C, H2P);
  }
  attn_dots<<<ngrid, TB, 0, stream>>>(h2p, aS2, aD2, as2, ad2, N, 1, OUT_C, H2P);
  edge_max<<<egrid, TB, 0, stream>>>(src, dst, as2, ad2, m2, E, 1);
  edge_expsum<<<egrid, TB, 0, stream>>>(src, dst, as2, ad2, m2, den2, E, 1);
  edge_agg<<<egrid, TB, 0, stream>>>(src, dst, as2, ad2, m2, den2, h2p, out,
                                     E, 1, OUT_C, H2P, OUT_C);
  add_bias<<<cdiv((long long)N * OUT_C, TB), TB, 0, stream>>>(out, b2, N * OUT_C, OUT_C);
}